// NeuralCDE_72404558676045
// MI455X (gfx1250) — compile-verified
//
#include <hip/hip_runtime.h>
#include <hip/hip_bf16.h>

typedef float v2f __attribute__((ext_vector_type(2)));
typedef float v8f __attribute__((ext_vector_type(8)));

namespace {
constexpr int kT    = 166;   // time steps
constexpr int kInCh = 8;     // input channels
constexpr int kHid  = 64;    // hidden
constexpr int kMlp  = 128;   // MLP width
constexpr int kOutW = 512;   // IN_CH*HID
constexpr int kMT   = 16;    // batch rows per workgroup

// LDS pitches (floats), chosen for conflict-free WMMA fragment gathers:
constexpr int PW1  = 136;    // W1 [64][128] (2*PW1 % 64 == 16)
constexpr int PW2P = 1056;   // W2 K-pair-interleaved: [64 pairs][512 cols x 2]
                             // (PW2P % 64 == 32 -> lane halves on disjoint banks)
constexpr int PH   = 132;    // H  [16][128]  (PH % 64 == 4)
constexpr int PZ   = 68;     // Z  [16][64]   (PZ % 64 == 4)
constexpr int PDX  = 9;      // dX [16][8]

constexpr int OFF_W1 = 0;
constexpr int OFF_W2 = OFF_W1 + kHid * PW1;       // 8704
constexpr int OFF_H  = OFF_W2 + (kMlp / 2) * PW2P;// 76288
constexpr int OFF_Z  = OFF_H  + kMT * PH;         // 78400
constexpr int OFF_DX = OFF_Z  + kMT * PZ;         // 79488
constexpr int OFF_B1 = OFF_DX + kMT * PDX;        // 79632
constexpr int OFF_B2 = OFF_B1 + kMlp;             // 79760
constexpr int OFF_WR = OFF_B2 + kOutW;            // 80272
constexpr int LDS_FLOATS = OFF_WR + kHid;         // 80336 -> 321344 B (< 320 KB)
} // namespace

__device__ __forceinline__ float fast_tanh(float x) {
#if __has_builtin(__builtin_amdgcn_tanhf)
    return __builtin_amdgcn_tanhf(x);        // native V_TANH_F32 on gfx1250
#else
    return 1.0f - 2.0f / (1.0f + __expf(2.0f * x));
#endif
}

__global__ __launch_bounds__(256, 1)
void ncde_persistent_kernel(const float* __restrict__ u0,
                            const float* __restrict__ coeffs,
                            const float* __restrict__ W1, const float* __restrict__ b1,
                            const float* __restrict__ W2, const float* __restrict__ b2,
                            const float* __restrict__ Wi, const float* __restrict__ bi,
                            const float* __restrict__ Wr, const float* __restrict__ br,
                            float* __restrict__ out)
{
    extern __shared__ float sm[];
    float* sW1 = sm + OFF_W1;
    float* sW2 = sm + OFF_W2;   // column-permuted (n' = (n&7)*64 + (n>>3)) AND
                                // K-pair interleaved: (k,n') -> (k>>1)*PW2P + 2n' + (k&1)
    float* sH  = sm + OFF_H;
    float* sZ  = sm + OFF_Z;
    float* sDX = sm + OFF_DX;
    float* sB1 = sm + OFF_B1;
    float* sB2 = sm + OFF_B2;   // column-permuted like sW2
    float* sWr = sm + OFF_WR;

    const int tid  = threadIdx.x;
    const int lane = tid & 31;
    const int w    = tid >> 5;        // wave id 0..7
    const int hi   = lane >> 4;       // lane half (K/M offset selector)
    const int l16  = lane & 15;
    const int row0 = blockIdx.x * kMT;

    // ---- stage weights & biases into LDS (once; reused 165x) ----
    for (int idx = tid; idx < kHid * kMlp; idx += 256)
        sW1[(idx >> 7) * PW1 + (idx & 127)] = W1[idx];
    for (int idx = tid; idx < kMlp * kOutW; idx += 256) {
        const int k = idx >> 9, n = idx & 511;
        const int nperm = (n & 7) * 64 + (n >> 3);     // i-major column order
        sW2[(k >> 1) * PW2P + 2 * nperm + (k & 1)] = W2[idx];
    }
    if (tid < kMlp) sB1[tid] = b1[tid];
    for (int idx = tid; idx < kOutW; idx += 256)
        sB2[(idx & 7) * 64 + (idx >> 3)] = b2[idx];
    if (tid < kHid) sWr[tid] = Wr[tid];

    // ---- dX software pipeline + dX(0) into LDS ----
    float cur = 0.f, inflight = 0.f;
    const float* cb = nullptr;
    if (tid < kMT * kInCh) {
        const int m = tid >> 3, i = tid & 7;
        cb = coeffs + (size_t)(row0 + m) * (kT * kInCh) + i;
        const float v0 = cb[0];
        cur = cb[kInCh];                      // val[1]
        sDX[m * PDX + i] = cur - v0;          // dX(0)
        inflight = cb[2 * kInCh];             // val[2] (in flight)
    }

    // ---- z0 = u0 @ Wi + bi (16 rows x 64 cols, 4 entries/thread) ----
#pragma unroll
    for (int r = 0; r < 4; ++r) {
        const int idx = tid + 256 * r;        // 0..1023
        const int m = idx >> 6, h = idx & 63;
        float s = bi[h];
        const float* up = u0 + (size_t)(row0 + m) * (kInCh - 1);
#pragma unroll
        for (int i = 0; i < kInCh - 1; ++i) s += up[i] * Wi[i * kHid + h];
        sZ[m * PZ + h] = s;
    }
    __syncthreads();

    // ---- hoist all loop-invariant LDS values into registers ----
    const int n1 = w * 16 + l16;              // GEMM1 column of this lane
    v2f wb1[16];                              // W1 B-fragments (invariant!)
#pragma unroll
    for (int k = 0; k < 16; ++k) {
        const int ka = 4 * k + 2 * hi;
        wb1[k].x = sW1[ka * PW1 + n1];
        wb1[k].y = sW1[(ka + 1) * PW1 + n1];
    }
    const float bias1 = sB1[n1];

    const int hg = w >> 1;                    // h-group 0..3 (16 h columns)
    const int ih = w & 1;                     // input-channel half (i = 4*ih + j)
    const int hcol = 16 * hg + l16;
    const int nb2 = 2 * (256 * ih + hcol);    // x2: K-pair-interleaved columns
    float bias2[4];
#pragma unroll
    for (int j = 0; j < 4; ++j) bias2[j] = sB2[(4 * ih + j) * 64 + hcol];

    const int mro = tid >> 4, sub = tid & 15; // readout mapping
    float wr4[4];
#pragma unroll
    for (int c = 0; c < 4; ++c) wr4[c] = sWr[sub * 4 + c];
    const float brv = br[0];

    // ---- readout t = 0 ----
    {
        float s = 0.f;
#pragma unroll
        for (int c = 0; c < 4; ++c) s += sZ[mro * PZ + sub * 4 + c] * wr4[c];
#pragma unroll
        for (int msk = 1; msk < 16; msk <<= 1) s += __shfl_xor(s, msk);
        if (sub == 0) out[(size_t)(row0 + mro) * kT + 0] = s + brv;
    }

    // ================= time loop =================
    for (int t = 0; t < kT - 1; ++t) {
        // ---- phase A: GEMM1 (H = relu(Z @ W1 + b1)); W1 frags in registers ----
        {
            v8f acc = {};
#pragma unroll
            for (int k = 0; k < 16; ++k) {
                const int ka = 4 * k + 2 * hi;
                v2f a = *(const v2f*)(sZ + l16 * PZ + ka);     // A frag: Z rows
                acc = __builtin_amdgcn_wmma_f32_16x16x4_f32(
                    false, a, false, wb1[k], (short)0, acc, false, false);
            }
#pragma unroll
            for (int v = 0; v < 8; ++v) {
                const int m = v + 8 * hi;
                sH[m * PH + n1] = fmaxf(acc[v] + bias1, 0.f);
            }
        }
        __syncthreads();

        // ---- phase B: issue coeffs[t+3] (wait hidden behind 128 WMMAs),
        //               GEMM2 on interleaved W2 + tanh + z += f . dX ----
        float newload = 0.f;
        if (t < kT - 3 && tid < kMT * kInCh)
            newload = cb[(size_t)(t + 3) * kInCh];
        {
            v8f acc[4] = {v8f{}, v8f{}, v8f{}, v8f{}};
#pragma unroll 4
            for (int k0 = 0; k0 < kMlp; k0 += 4) {
                const int ka = k0 + 2 * hi;                    // even
                v2f a = *(const v2f*)(sH + l16 * PH + ka);
                const float* bp = sW2 + (ka >> 1) * PW2P + nb2;
#pragma unroll
                for (int j = 0; j < 4; ++j) {                  // 4 i-tiles
                    v2f b = *(const v2f*)(bp + 128 * j);       // one b64 load
                    acc[j] = __builtin_amdgcn_wmma_f32_16x16x4_f32(
                        false, a, false, b, (short)0, acc[j], false, false);
                }
            }
            // epilogue: reduction over i is IN-LANE (4 FMAs), then one LDS
            // float atomic combines the two wave-halves into z.
#pragma unroll
            for (int v = 0; v < 8; ++v) {
                const int m = v + 8 * hi;
                float s = 0.f;
#pragma unroll
                for (int j = 0; j < 4; ++j) {
                    const float f = fast_tanh(acc[j][v] + bias2[j]);
                    s += f * sDX[m * PDX + 4 * ih + j];
                }
                atomicAdd(&sZ[m * PZ + hcol], s); // ds_add_f32 (no return)
            }
        }
        __syncthreads();

        // ---- phase C: readout t+1 + store dX(t+1), rotate prefetch regs ----
        {
            float s = 0.f;
#pragma unroll
            for (int c = 0; c < 4; ++c) s += sZ[mro * PZ + sub * 4 + c] * wr4[c];
#pragma unroll
            for (int msk = 1; msk < 16; msk <<= 1) s += __shfl_xor(s, msk);
            if (sub == 0) out[(size_t)(row0 + mro) * kT + (t + 1)] = s + brv;
        }
        if (tid < kMT * kInCh) {                  // consumed after next B1 barrier
            sDX[(tid >> 3) * PDX + (tid & 7)] = inflight - cur;
            cur = inflight;
            inflight = newload;                   // issued in phase B(t)
        }
        // no barrier needed: next phase A only reads sZ/sW1 and writes sH,
        // none of which phase C touches; sDX readers are behind barrier B1.
    }
}

extern "C" void kernel_launch(void* const* d_in, const int* in_sizes, int n_in,
                              void* d_out, int out_size, void* d_ws, size_t ws_size,
                              hipStream_t stream) {
    (void)in_sizes; (void)n_in; (void)d_ws; (void)ws_size; (void)out_size;
    const float* u0     = (const float*)d_in[0];
    const float* coeffs = (const float*)d_in[1];
    const float* W1     = (const float*)d_in[2];
    const float* b1     = (const float*)d_in[3];
    const float* W2     = (const float*)d_in[4];
    const float* b2     = (const float*)d_in[5];
    const float* Wi     = (const float*)d_in[6];
    const float* bi     = (const float*)d_in[7];
    const float* Wr     = (const float*)d_in[8];
    const float* br     = (const float*)d_in[9];
    float* out = (float*)d_out;

    const size_t lds_bytes = (size_t)LDS_FLOATS * sizeof(float);
    hipFuncSetAttribute((const void*)ncde_persistent_kernel,
                        hipFuncAttributeMaxDynamicSharedMemorySize, (int)lds_bytes);

    const int n_blocks = 4096 / kMT;   // 256 workgroups, 16 batch rows each
    ncde_persistent_kernel<<<n_blocks, 256, lds_bytes, stream>>>(
        u0, coeffs, W1, b1, W2, b2, Wi, bi, Wr, br, out);
}